// MyMultiheadAttentionCon_14482629722604
// MI455X (gfx1250) — compile-verified
//
#include <hip/hip_runtime.h>
#include <math.h>

typedef __attribute__((ext_vector_type(2))) float v2f;
typedef __attribute__((ext_vector_type(8))) float v8f;

#define D_MODEL 1024
#define NUM_HEADS 16
#define HEAD 64
#define BATCH 2
#define SEQ 2048
#define ROWS (BATCH * SEQ)   /* 4096 */
#define KTILES (SEQ / 16)    /* 128 */

__device__ __forceinline__ v8f wmma_f32(v2f a, v2f b, v8f c) {
    // V_WMMA_F32_16X16X4_F32: fp32 A(16x4), B(4x16), C/D(16x16) — full fp32 path
    return __builtin_amdgcn_wmma_f32_16x16x4_f32(
        /*neg_a=*/false, a, /*neg_b=*/false, b,
        /*c_mod=*/(short)0, c, /*reuse_a=*/false, /*reuse_b=*/false);
}

// ---------------------------------------------------------------------------
// Kernel 1: Y = X @ W^T + b, stored as [B, H, S, 64].
// One wave computes a 16x64 strip (4 WMMA accumulators). blockIdx.z picks Q/K/V.
// ---------------------------------------------------------------------------
__global__ __launch_bounds__(256)
void proj_kernel(const float* __restrict__ Xq, const float* __restrict__ Xk,
                 const float* __restrict__ Xv,
                 const float* __restrict__ Wq, const float* __restrict__ bq,
                 const float* __restrict__ Wk, const float* __restrict__ bk,
                 const float* __restrict__ Wv, const float* __restrict__ bv,
                 float* __restrict__ Qo, float* __restrict__ Ko,
                 float* __restrict__ Vo)
{
    const float* X; const float* W; const float* bias; float* Y;
    if (blockIdx.z == 0)      { X = Xq; W = Wq; bias = bq; Y = Qo; }
    else if (blockIdx.z == 1) { X = Xk; W = Wk; bias = bk; Y = Ko; }
    else                      { X = Xv; W = Wv; bias = bv; Y = Vo; }

    const int wid  = blockIdx.x * 8 + (threadIdx.x >> 5); // 0..4095
    const int lane = threadIdx.x & 31;
    const int half = lane >> 4;      // K-slice group of the fp32 A/B layout
    const int lm   = lane & 15;      // M for A, N for B/C
    const int mt   = wid >> 4;       // 0..255 (16-row tile)
    const int gh   = wid & 15;       // 64-col group == head index

    const float* xrow = X + (mt * 16 + lm) * D_MODEL + half * 2;
    const float* wr0  = W + (gh * 64 +  0 + lm) * D_MODEL + half * 2;
    const float* wr1  = W + (gh * 64 + 16 + lm) * D_MODEL + half * 2;
    const float* wr2  = W + (gh * 64 + 32 + lm) * D_MODEL + half * 2;
    const float* wr3  = W + (gh * 64 + 48 + lm) * D_MODEL + half * 2;

    v8f acc[4] = {};
#pragma unroll 4
    for (int k0 = 0; k0 < D_MODEL; k0 += 4) {
        v2f a = *(const v2f*)(xrow + k0);
        acc[0] = wmma_f32(a, *(const v2f*)(wr0 + k0), acc[0]);
        acc[1] = wmma_f32(a, *(const v2f*)(wr1 + k0), acc[1]);
        acc[2] = wmma_f32(a, *(const v2f*)(wr2 + k0), acc[2]);
        acc[3] = wmma_f32(a, *(const v2f*)(wr3 + k0), acc[3]);
    }

#pragma unroll
    for (int nt = 0; nt < 4; ++nt) {
        const float bval = bias[gh * 64 + nt * 16 + lm];
#pragma unroll
        for (int v = 0; v < 8; ++v) {
            const int m = mt * 16 + half * 8 + v;   // global row in [0,4096)
            const int b = m >> 11;                  // batch
            const int s = m & (SEQ - 1);            // sequence pos
            Y[((b * NUM_HEADS + gh) * SEQ + s) * HEAD + nt * 16 + lm] =
                acc[nt][v] + bval;
        }
    }
}

// ---------------------------------------------------------------------------
// Kernel 2: flash-style attention with reciprocal scores. One wave per
// (b, h, q-tile). Saves row max/sum (m, l) for the attn_mean pass.
// ---------------------------------------------------------------------------
__global__ __launch_bounds__(256)
void attn_kernel(const float* __restrict__ Q, const float* __restrict__ K,
                 const float* __restrict__ V, float* __restrict__ out,
                 float* __restrict__ mbuf, float* __restrict__ lbuf)
{
    __shared__ float plds[8 * 256];   // one 16x16 P tile per wave

    const int wib  = threadIdx.x >> 5;
    const int wid  = blockIdx.x * 8 + wib;   // 0..4095
    const int lane = threadIdx.x & 31;
    const int half = lane >> 4;
    const int lm   = lane & 15;
    const int b    = wid >> 11;
    const int h    = (wid >> 7) & 15;
    const int qt   = wid & 127;
    const int bh   = b * NUM_HEADS + h;

    // Cache the 16x64 Q tile as 16 A-fragments (fp32 A layout).
    const float* qrow = Q + (bh * SEQ + qt * 16 + lm) * HEAD + half * 2;
    v2f qf[16];
#pragma unroll
    for (int i = 0; i < 16; ++i) qf[i] = *(const v2f*)(qrow + i * 4);

    v8f   O[4] = {};
    float mrow[8], lrow[8];
#pragma unroll
    for (int v = 0; v < 8; ++v) { mrow[v] = -INFINITY; lrow[v] = 0.0f; }

    float* pw = plds + wib * 256;

    for (int kt = 0; kt < KTILES; ++kt) {
        const float* krow = K + (bh * SEQ + kt * 16 + lm) * HEAD + half * 2;
        v8f raw = {};
#pragma unroll
        for (int i = 0; i < 16; ++i)
            raw = wmma_f32(qf[i], *(const v2f*)(krow + i * 4), raw);

        // scores = (1/raw)/sqrt(64); online softmax per row (rows of C layout
        // live across the 16-lane half-wave -> width-16 shuffle reductions).
#pragma unroll
        for (int v = 0; v < 8; ++v) {
            float s = 1.0f / (8.0f * raw[v]);
            float t = s;
            t = fmaxf(t, __shfl_xor(t, 1, 16));
            t = fmaxf(t, __shfl_xor(t, 2, 16));
            t = fmaxf(t, __shfl_xor(t, 4, 16));
            t = fmaxf(t, __shfl_xor(t, 8, 16));
            const float mn = fmaxf(mrow[v], t);
            const float sc = __expf(mrow[v] - mn);
            const float e  = __expf(s - mn);
            float es = e;
            es += __shfl_xor(es, 1, 16);
            es += __shfl_xor(es, 2, 16);
            es += __shfl_xor(es, 4, 16);
            es += __shfl_xor(es, 8, 16);
            lrow[v] = lrow[v] * sc + es;
            mrow[v] = mn;
#pragma unroll
            for (int nt = 0; nt < 4; ++nt) O[nt][v] *= sc;
            pw[(half * 8 + v) * 16 + lm] = e;     // C layout -> LDS
        }
        // Same-wave LDS ops are in-order; just drain before A-layout re-read.
        asm volatile("s_wait_dscnt 0" ::: "memory");

        // O += P(16x16) @ V(16x64): re-read P in A layout, V as B fragments.
        const float* vbase = V + (bh * SEQ + kt * 16) * HEAD;
#pragma unroll
        for (int kk = 0; kk < 4; ++kk) {
            v2f af = *(const v2f*)(pw + lm * 16 + kk * 4 + half * 2);
#pragma unroll
            for (int nt = 0; nt < 4; ++nt) {
                v2f bf;
                bf[0] = vbase[(kk * 4 + half * 2 + 0) * HEAD + nt * 16 + lm];
                bf[1] = vbase[(kk * 4 + half * 2 + 1) * HEAD + nt * 16 + lm];
                O[nt] = wmma_f32(af, bf, O[nt]);
            }
        }
    }

#pragma unroll
    for (int v = 0; v < 8; ++v) {
        const int s   = qt * 16 + half * 8 + v;
        const float r = 1.0f / lrow[v];
#pragma unroll
        for (int nt = 0; nt < 4; ++nt)
            out[(b * SEQ + s) * D_MODEL + h * HEAD + nt * 16 + lm] = O[nt][v] * r;
        if (lm == 0) {
            mbuf[bh * SEQ + s] = mrow[v];
            lbuf[bh * SEQ + s] = lrow[v];
        }
    }
}

// ---------------------------------------------------------------------------
// Kernel 3: attn_mean = mean over heads of exp(1/(8*raw) - m)/l.
// One wave per (b, q-tile, k-tile); recomputes QK^T per head (compute is cheap).
// ---------------------------------------------------------------------------
__global__ __launch_bounds__(256)
void mean_kernel(const float* __restrict__ Q, const float* __restrict__ K,
                 const float* __restrict__ mbuf, const float* __restrict__ lbuf,
                 float* __restrict__ attn_mean)
{
    const int wid  = blockIdx.x * 8 + (threadIdx.x >> 5);  // 0..32767
    const int lane = threadIdx.x & 31;
    const int half = lane >> 4;
    const int lm   = lane & 15;
    const int b    = wid >> 14;
    const int qt   = (wid >> 7) & 127;
    const int kt   = wid & 127;

    v8f acc = {};
    for (int h = 0; h < NUM_HEADS; ++h) {
        const int bh = b * NUM_HEADS + h;
        const float* qrow = Q + (bh * SEQ + qt * 16 + lm) * HEAD + half * 2;
        const float* krow = K + (bh * SEQ + kt * 16 + lm) * HEAD + half * 2;
        v8f raw = {};
#pragma unroll
        for (int i = 0; i < 16; ++i)
            raw = wmma_f32(*(const v2f*)(qrow + i * 4),
                           *(const v2f*)(krow + i * 4), raw);
#pragma unroll
        for (int v = 0; v < 8; ++v) {
            const int s    = qt * 16 + half * 8 + v;
            const float mh = mbuf[bh * SEQ + s];
            const float lh = lbuf[bh * SEQ + s];
            const float sc = 1.0f / (8.0f * raw[v]);
            acc[v] += __expf(sc - mh) / lh;
        }
    }
#pragma unroll
    for (int v = 0; v < 8; ++v) {
        const int s = qt * 16 + half * 8 + v;
        attn_mean[(b * SEQ + s) * SEQ + kt * 16 + lm] = acc[v] * (1.0f / 16.0f);
    }
}

// ---------------------------------------------------------------------------
extern "C" void kernel_launch(void* const* d_in, const int* in_sizes, int n_in,
                              void* d_out, int out_size, void* d_ws, size_t ws_size,
                              hipStream_t stream)
{
    const float* query = (const float*)d_in[0];
    const float* key   = (const float*)d_in[1];
    const float* value = (const float*)d_in[2];
    const float* Wq    = (const float*)d_in[3];
    const float* bq    = (const float*)d_in[4];
    const float* Wk    = (const float*)d_in[5];
    const float* bk    = (const float*)d_in[6];
    const float* Wv    = (const float*)d_in[7];
    const float* bv    = (const float*)d_in[8];

    float* out       = (float*)d_out;                 // [B,S,D_MODEL]
    float* attn_mean = out + (size_t)ROWS * D_MODEL;  // [B,S,S]

    float* ws   = (float*)d_ws;
    float* Qp   = ws;                                 // [B,H,S,64]
    float* Kp   = Qp + (size_t)ROWS * D_MODEL;
    float* Vp   = Kp + (size_t)ROWS * D_MODEL;
    float* mbuf = Vp + (size_t)ROWS * D_MODEL;        // [B,H,S]
    float* lbuf = mbuf + (size_t)BATCH * NUM_HEADS * SEQ;

    // 4096 waves per projection, 8 waves/block; z selects Q/K/V.
    proj_kernel<<<dim3(512, 1, 3), 256, 0, stream>>>(
        query, key, value, Wq, bq, Wk, bk, Wv, bv, Qp, Kp, Vp);

    // 2*16*128 = 4096 waves.
    attn_kernel<<<512, 256, 0, stream>>>(Qp, Kp, Vp, out, mbuf, lbuf);

    // 2*128*128 = 32768 waves.
    mean_kernel<<<4096, 256, 0, stream>>>(Qp, Kp, mbuf, lbuf, attn_mean);
}